// Model_41798621725133
// MI455X (gfx1250) — compile-verified
//
#include <hip/hip_runtime.h>
#include <math.h>

typedef float v2f __attribute__((ext_vector_type(2)));
typedef float v8f __attribute__((ext_vector_type(8)));

// ---- problem sizes ----
constexpr int Bsz = 4096, NG = 1440, Dd = 50, DRn = 222, NAn = 64, LGn = 3;
constexpr int C1 = 10, L1 = 713;          // conv1: (1440-15)/2+1
constexpr int C2 = 5,  L2 = 350;          // conv2: (713-15)/2+1
constexpr int LP = 70, GO = 32;           // maxpool5 -> 70 ; gene linear out 32
constexpr int CONC = 210, MH = 100;       // concat 160+50 ; mlin out
constexpr int ML1 = 46, MP1 = 15, ML2 = 3;// mconv1 len, pool3 len, mconv2 len

__device__ __forceinline__ float block_reduce_sum(float v, float* red, int t, int n) {
  red[t] = v; __syncthreads();
  for (int s = n >> 1; s > 0; s >>= 1) {
    if (t < s) red[t] += red[t + s];
    __syncthreads();
  }
  float r = red[0]; __syncthreads();
  return r;
}

__device__ __forceinline__ float clip01(float x) { return fminf(fmaxf(x, 0.f), 1.f); }

// ---- P = sim_softmax (222x222) @ clip(fuse_weight) (222x1440), f32 WMMA ----
// One wave per 16x16 tile. Unconditional clamped loads + mask multiplies (no
// divergent predicated loads in the K loop); K remainder peeled.
__global__ void wmma_P_kernel(const float* __restrict__ S, const float* __restrict__ FW,
                              float* __restrict__ P) {
  int n0 = blockIdx.x * 16, m0 = blockIdx.y * 16;
  int lane = threadIdx.x, r = lane & 15, h = lane >> 4;
  int row = m0 + r;
  int rowc = row < DRn ? row : (DRn - 1);
  float rm = row < DRn ? 1.f : 0.f;           // loop-invariant row mask
  const float* Srow = S + (size_t)rowc * DRn;
  const float* Bcol = FW + n0 + r;
  v8f acc = {0.f,0.f,0.f,0.f,0.f,0.f,0.f,0.f};
  // main loop: k in [0,219] all valid (222 = 55*4 + 2)
  for (int kk = 0; kk < 220; kk += 4) {
    int ka = kk + 2 * h;
    v2f a, b;
    a.x = Srow[ka] * rm;
    a.y = Srow[ka + 1] * rm;
    b.x = clip01(Bcol[(size_t)ka * NG]);
    b.y = clip01(Bcol[(size_t)(ka + 1) * NG]);
    acc = __builtin_amdgcn_wmma_f32_16x16x4_f32(false, a, false, b, (short)0, acc, false, false);
  }
  { // peeled remainder: kk=220 covers k=220..223, valid k=220,221
    int ka = 220 + 2 * h;
    float k0 = (ka     < DRn) ? 1.f : 0.f;
    float k1 = (ka + 1 < DRn) ? 1.f : 0.f;
    int ka0 = (ka     < DRn) ? ka     : (DRn - 1);
    int ka1 = (ka + 1 < DRn) ? ka + 1 : (DRn - 1);
    v2f a, b;
    a.x = Srow[ka0] * (rm * k0);
    a.y = Srow[ka1] * (rm * k1);
    b.x = clip01(Bcol[(size_t)ka0 * NG]);
    b.y = clip01(Bcol[(size_t)ka1 * NG]);
    acc = __builtin_amdgcn_wmma_f32_16x16x4_f32(false, a, false, b, (short)0, acc, false, false);
  }
#pragma unroll
  for (int i = 0; i < 8; i++) {
    int orow = m0 + i + 8 * h, col = n0 + r;
    if (orow < DRn) P[(size_t)orow * NG + col] = acc[i];
  }
}

// ---- m = tanh(concat (4096x210) @ mlin_w (210x100) + b), f32 WMMA ----
__global__ void wmma_mlin_kernel(const float* __restrict__ X, const float* __restrict__ W,
                                 const float* __restrict__ bias, float* __restrict__ Mout) {
  int n0 = blockIdx.x * 16, m0 = blockIdx.y * 16;
  int lane = threadIdx.x, r = lane & 15, h = lane >> 4;
  int col = n0 + r;
  int colc = col < MH ? col : (MH - 1);
  float cm = col < MH ? 1.f : 0.f;            // loop-invariant col mask
  const float* Arow = X + (size_t)(m0 + r) * CONC;  // rows always valid (4096 % 16 == 0)
  const float* Bc = W + colc;
  v8f acc = {0.f,0.f,0.f,0.f,0.f,0.f,0.f,0.f};
  // main loop: k in [0,207] all valid (210 = 52*4 + 2)
  for (int kk = 0; kk < 208; kk += 4) {
    int ka = kk + 2 * h;
    v2f a, b;
    a.x = Arow[ka];
    a.y = Arow[ka + 1];
    b.x = Bc[(size_t)ka * MH] * cm;
    b.y = Bc[(size_t)(ka + 1) * MH] * cm;
    acc = __builtin_amdgcn_wmma_f32_16x16x4_f32(false, a, false, b, (short)0, acc, false, false);
  }
  { // peeled remainder: kk=208 covers k=208..211, valid k=208,209
    int ka = 208 + 2 * h;
    float k0 = (ka     < CONC) ? 1.f : 0.f;
    float k1 = (ka + 1 < CONC) ? 1.f : 0.f;
    int ka0 = (ka     < CONC) ? ka     : (CONC - 1);
    int ka1 = (ka + 1 < CONC) ? ka + 1 : (CONC - 1);
    v2f a, b;
    a.x = Arow[ka0] * k0;
    a.y = Arow[ka1] * k1;
    b.x = Bc[(size_t)ka0 * MH] * cm;
    b.y = Bc[(size_t)ka1 * MH] * cm;
    acc = __builtin_amdgcn_wmma_f32_16x16x4_f32(false, a, false, b, (short)0, acc, false, false);
  }
  if (col < MH) {
    float bs = bias[col];
#pragma unroll
    for (int i = 0; i < 8; i++) {
      int orow = m0 + i + 8 * h;
      Mout[(size_t)orow * MH + col] = tanhf(acc[i] + bs);
    }
  }
}

// ---- per-drug GNN: xs=embed[fp]; 3x {hs=relu(xs@W+b); xs+=A@hs}; mean ----
__global__ void gnn_kernel(const float* __restrict__ embed, const float* __restrict__ gW,
                           const float* __restrict__ gb, const int* __restrict__ fps,
                           const float* __restrict__ adj, float* __restrict__ comp) {
  __shared__ float xs[NAn * Dd];
  __shared__ float hs[NAn * Dd];
  __shared__ float Am[NAn * NAn];
  __shared__ float Wm[Dd * Dd];
  __shared__ float bb[Dd];
  __shared__ int fp[NAn];
  int d = blockIdx.x, t = threadIdx.x;
  if (t < NAn) fp[t] = fps[d * NAn + t];
  __syncthreads();
  for (int i = t; i < NAn * Dd; i += 256) { int n = i / Dd, e = i - n * Dd; xs[i] = embed[(size_t)fp[n] * Dd + e]; }
  for (int i = t; i < NAn * NAn; i += 256) Am[i] = adj[(size_t)d * NAn * NAn + i];
  __syncthreads();
  for (int layer = 0; layer < LGn; layer++) {
    for (int i = t; i < Dd * Dd; i += 256) Wm[i] = gW[layer * Dd * Dd + i];
    if (t < Dd) bb[t] = gb[layer * Dd + t];
    __syncthreads();
    for (int i = t; i < NAn * Dd; i += 256) {
      int n = i / Dd, e = i - n * Dd;
      float acc = bb[e];
      for (int k = 0; k < Dd; k++) acc += xs[n * Dd + k] * Wm[k * Dd + e];
      hs[i] = acc > 0.f ? acc : 0.f;
    }
    __syncthreads();
    for (int i = t; i < NAn * Dd; i += 256) {
      int n = i / Dd, e = i - n * Dd;
      float acc = 0.f;
      for (int m = 0; m < NAn; m++) acc += Am[n * NAn + m] * hs[m * Dd + e];
      xs[i] += acc;
    }
    __syncthreads();
  }
  if (t < Dd) {
    float s = 0.f;
    for (int n = 0; n < NAn; n++) s += xs[n * Dd + t];
    comp[d * Dd + t] = s * (1.0f / NAn);
  }
}

// ---- pass 1: com = rma + P[did]*var (store), conv1 stats only ----
__global__ void conv1_stats_kernel(const float* __restrict__ rma, const float* __restrict__ var,
                                   const int* __restrict__ drug_id, const float* __restrict__ P,
                                   const float* __restrict__ w1, const float* __restrict__ b1,
                                   float* __restrict__ comOut, float* __restrict__ partial) {
  __shared__ float sc[NG];
  __shared__ float w1s[C1 * 15];
  __shared__ float red[256];
  int b = blockIdx.x, t = threadIdx.x;
  int did = drug_id[b];
  for (int g = t; g < NG; g += 256) {
    float cm = rma[(size_t)b * NG + g] + P[(size_t)did * NG + g] * var[(size_t)b * NG + g];
    sc[g] = cm;
    comOut[(size_t)b * NG + g] = cm;
  }
  for (int i = t; i < C1 * 15; i += 256) w1s[i] = w1[i];
  __syncthreads();
  for (int c = 0; c < C1; c++) {
    float bc = b1[c], s = 0.f, ss = 0.f;
    for (int l = t; l < L1; l += 256) {
      float acc = bc; int base = 2 * l;
#pragma unroll
      for (int k = 0; k < 15; k++) acc += sc[base + k] * w1s[c * 15 + k];
      s += acc; ss += acc * acc;
    }
    float S = block_reduce_sum(s, red, t, 256);
    float SS = block_reduce_sum(ss, red, t, 256);
    if (t == 0) { partial[(b * C1 + c) * 2] = S; partial[(b * C1 + c) * 2 + 1] = SS; }
  }
}

// ---- per-channel BN params from partial sums: ab[c]=scale, ab[C+c]=shift ----
__global__ void bn_reduce_kernel(const float* __restrict__ partial, int nblk, int C, float invN,
                                 const float* __restrict__ g, const float* __restrict__ beta,
                                 float* __restrict__ ab) {
  int c = threadIdx.x;
  if (c >= C) return;
  float s = 0.f, ss = 0.f;
  for (int i = 0; i < nblk; i++) { s += partial[(i * C + c) * 2]; ss += partial[(i * C + c) * 2 + 1]; }
  float m = s * invN;
  float v = ss * invN - m * m;
  float a = g[c] * rsqrtf(v + 1e-5f);
  ab[c] = a; ab[C + c] = beta[c] - m * a;
}

// ---- pass 2: recompute conv1 from com, apply bn1+relu, conv2 (store) + stats ----
__global__ void conv2_stats_kernel(const float* __restrict__ com, const float* __restrict__ w1,
                                   const float* __restrict__ b1, const float* __restrict__ ab1,
                                   const float* __restrict__ w2, const float* __restrict__ b2,
                                   float* __restrict__ out2, float* __restrict__ partial) {
  __shared__ float sc[NG];
  __shared__ float y1[C1 * L1];
  __shared__ float w1s[C1 * 15];
  __shared__ float w2s[C2 * C1 * 15];
  __shared__ float red[256];
  int b = blockIdx.x, t = threadIdx.x;
  for (int g = t; g < NG; g += 256) sc[g] = com[(size_t)b * NG + g];
  for (int i = t; i < C1 * 15; i += 256) w1s[i] = w1[i];
  for (int i = t; i < C2 * C1 * 15; i += 256) w2s[i] = w2[i];
  __syncthreads();
  for (int idx = t; idx < C1 * L1; idx += 256) {
    int c = idx / L1, l = idx - c * L1;
    float acc = b1[c]; int base = 2 * l;
#pragma unroll
    for (int k = 0; k < 15; k++) acc += sc[base + k] * w1s[c * 15 + k];
    float v = ab1[c] * acc + ab1[C1 + c];
    y1[idx] = v > 0.f ? v : 0.f;
  }
  __syncthreads();
  for (int c = 0; c < C2; c++) {
    float bc = b2[c], s = 0.f, ss = 0.f;
    for (int l = t; l < L2; l += 256) {
      float acc = bc; int base = 2 * l;
      for (int ci = 0; ci < C1; ci++) {
#pragma unroll
        for (int k = 0; k < 15; k++) acc += y1[ci * L1 + base + k] * w2s[(c * C1 + ci) * 15 + k];
      }
      out2[((size_t)b * C2 + c) * L2 + l] = acc;
      s += acc; ss += acc * acc;
    }
    float S = block_reduce_sum(s, red, t, 256);
    float SS = block_reduce_sum(ss, red, t, 256);
    if (t == 0) { partial[(b * C2 + c) * 2] = S; partial[(b * C2 + c) * 2 + 1] = SS; }
  }
}

// ---- bn2+relu, maxpool5, gene linear+relu, write concat row (+compound gather) ----
__global__ void gene_final_kernel(const float* __restrict__ out2, const float* __restrict__ ab2,
                                  const float* __restrict__ Wg, const float* __restrict__ bg,
                                  const float* __restrict__ comp, const int* __restrict__ drug_id,
                                  float* __restrict__ conc) {
  __shared__ float y[C2 * L2];
  __shared__ float pooled[C2 * LP];
  int b = blockIdx.x, t = threadIdx.x;
  for (int i = t; i < C2 * L2; i += 256) {
    int c = i / L2;
    float v = ab2[c] * out2[(size_t)b * C2 * L2 + i] + ab2[C2 + c];
    y[i] = v > 0.f ? v : 0.f;
  }
  __syncthreads();
  for (int i = t; i < C2 * LP; i += 256) {
    int c = i / LP, j = i - c * LP;
    float mx = y[c * L2 + j * 5];
#pragma unroll
    for (int k = 1; k < 5; k++) mx = fmaxf(mx, y[c * L2 + j * 5 + k]);
    pooled[i] = mx;
  }
  __syncthreads();
  for (int i = t; i < C2 * GO; i += 256) {
    int c = i / GO, o = i - c * GO;
    float acc = bg[o];
    for (int l = 0; l < LP; l++) acc += pooled[c * LP + l] * Wg[l * GO + o];
    conc[(size_t)b * CONC + i] = acc > 0.f ? acc : 0.f;
  }
  int did = drug_id[b];
  for (int i = t; i < Dd; i += 256) conc[(size_t)b * CONC + C2 * GO + i] = comp[did * Dd + i];
}

// ---- mconv1 (store raw) + stats ----
__global__ void mconv1_stats_kernel(const float* __restrict__ M, const float* __restrict__ w,
                                    const float* __restrict__ bias, float* __restrict__ raw,
                                    float* __restrict__ partial) {
  __shared__ float sm[MH];
  __shared__ float red[64];
  int b = blockIdx.x, t = threadIdx.x;
  for (int i = t; i < MH; i += 64) sm[i] = M[(size_t)b * MH + i];
  __syncthreads();
  for (int c = 0; c < C2; c++) {
    float bc = bias[c], s = 0.f, ss = 0.f;
    for (int l = t; l < ML1; l += 64) {
      float acc = bc;
#pragma unroll
      for (int k = 0; k < 10; k++) acc += sm[2 * l + k] * w[c * 10 + k];
      raw[((size_t)b * C2 + c) * ML1 + l] = acc;
      s += acc; ss += acc * acc;
    }
    float S = block_reduce_sum(s, red, t, 64);
    float SS = block_reduce_sum(ss, red, t, 64);
    if (t == 0) { partial[(b * C2 + c) * 2] = S; partial[(b * C2 + c) * 2 + 1] = SS; }
  }
}

// ---- apply mbn1 (no relu), maxpool3, mconv2 (store raw) + stats ----
__global__ void mconv2_stats_kernel(const float* __restrict__ raw1, const float* __restrict__ ab,
                                    const float* __restrict__ w, const float* __restrict__ bias,
                                    float* __restrict__ raw2, float* __restrict__ partial) {
  __shared__ float y[C2 * ML1];
  __shared__ float pooled[C2 * MP1];
  __shared__ float sv[16], sq[16];
  int b = blockIdx.x, t = threadIdx.x;
  for (int i = t; i < C2 * ML1; i += 64) {
    int c = i / ML1;
    y[i] = ab[c] * raw1[(size_t)b * C2 * ML1 + i] + ab[C2 + c];
  }
  __syncthreads();
  for (int i = t; i < C2 * MP1; i += 64) {
    int c = i / MP1, j = i - c * MP1;
    float mx = fmaxf(y[c * ML1 + 3 * j], fmaxf(y[c * ML1 + 3 * j + 1], y[c * ML1 + 3 * j + 2]));
    pooled[i] = mx;
  }
  __syncthreads();
  if (t < C2 * ML2) {
    int c = t / ML2, l = t - c * ML2;
    float acc = bias[c];
    for (int ci = 0; ci < C2; ci++) {
#pragma unroll
      for (int k = 0; k < 10; k++) acc += pooled[ci * MP1 + 2 * l + k] * w[(c * C2 + ci) * 10 + k];
    }
    raw2[(size_t)b * C2 * ML2 + t] = acc;
    sv[t] = acc; sq[t] = acc * acc;
  }
  __syncthreads();
  if (t < C2) {
    float s = sv[t * 3] + sv[t * 3 + 1] + sv[t * 3 + 2];
    float ss = sq[t * 3] + sq[t * 3 + 1] + sq[t * 3 + 2];
    partial[(b * C2 + t) * 2] = s; partial[(b * C2 + t) * 2 + 1] = ss;
  }
}

// ---- apply mbn2, maxpool3 (->1), final dot with out_w ----
__global__ void final_kernel(const float* __restrict__ raw2, const float* __restrict__ ab,
                             const float* __restrict__ ow, const float* __restrict__ ob,
                             float* __restrict__ out) {
  int i = blockIdx.x * blockDim.x + threadIdx.x;
  if (i >= Bsz) return;
  float acc = ob[0];
  for (int c = 0; c < C2; c++) {
    float a = ab[c], sh = ab[C2 + c];
    float mx = a * raw2[(size_t)i * C2 * ML2 + c * ML2] + sh;
#pragma unroll
    for (int l = 1; l < ML2; l++) mx = fmaxf(mx, a * raw2[(size_t)i * C2 * ML2 + c * ML2 + l] + sh);
    acc += mx * ow[c];
  }
  out[i] = acc;
}

extern "C" void kernel_launch(void* const* d_in, const int* in_sizes, int n_in,
                              void* d_out, int out_size, void* d_ws, size_t ws_size,
                              hipStream_t stream) {
  (void)in_sizes; (void)n_in; (void)out_size; (void)ws_size;
  const float* rma          = (const float*)d_in[0];
  const float* var          = (const float*)d_in[1];
  const int*   drug_id      = (const int*)  d_in[2];
  const float* fuse_weight  = (const float*)d_in[3];
  const float* sim_softmax  = (const float*)d_in[4];
  const float* embed_table  = (const float*)d_in[5];
  const float* gnn_W        = (const float*)d_in[6];
  const float* gnn_b        = (const float*)d_in[7];
  const int*   fingerprints = (const int*)  d_in[8];
  const float* adjacency    = (const float*)d_in[9];
  const float* conv1_w = (const float*)d_in[10]; const float* conv1_b = (const float*)d_in[11];
  const float* bn1_g   = (const float*)d_in[12]; const float* bn1_b   = (const float*)d_in[13];
  const float* conv2_w = (const float*)d_in[14]; const float* conv2_b = (const float*)d_in[15];
  const float* bn2_g   = (const float*)d_in[16]; const float* bn2_b   = (const float*)d_in[17];
  const float* lin_gene_w = (const float*)d_in[18]; const float* lin_gene_b = (const float*)d_in[19];
  const float* mlin_w  = (const float*)d_in[20]; const float* mlin_b  = (const float*)d_in[21];
  const float* mconv1_w = (const float*)d_in[22]; const float* mconv1_b = (const float*)d_in[23];
  const float* mbn1_g  = (const float*)d_in[24]; const float* mbn1_b  = (const float*)d_in[25];
  const float* mconv2_w = (const float*)d_in[26]; const float* mconv2_b = (const float*)d_in[27];
  const float* mbn2_g  = (const float*)d_in[28]; const float* mbn2_b  = (const float*)d_in[29];
  const float* out_w   = (const float*)d_in[30]; const float* out_b   = (const float*)d_in[31];

  float* ws = (float*)d_ws;
  size_t off = 0;
  float* P    = ws + off; off += (size_t)DRn * NG;
  float* com  = ws + off; off += (size_t)Bsz * NG;
  float* out2 = ws + off; off += (size_t)Bsz * C2 * L2;
  float* p1   = ws + off; off += (size_t)Bsz * C1 * 2;
  float* ab1  = ws + off; off += 2 * C1;
  float* p2   = ws + off; off += (size_t)Bsz * C2 * 2;
  float* ab2  = ws + off; off += 2 * C2;
  float* conc = ws + off; off += (size_t)Bsz * CONC;
  float* comp = ws + off; off += (size_t)DRn * Dd;
  float* Mt   = ws + off; off += (size_t)Bsz * MH;
  float* m1   = ws + off; off += (size_t)Bsz * C2 * ML1;
  float* pm1  = ws + off; off += (size_t)Bsz * C2 * 2;
  float* abm1 = ws + off; off += 2 * C2;
  float* m2   = ws + off; off += (size_t)Bsz * C2 * ML2;
  float* pm2  = ws + off; off += (size_t)Bsz * C2 * 2;
  float* abm2 = ws + off; off += 2 * C2;

  wmma_P_kernel<<<dim3(NG / 16, (DRn + 15) / 16), 32, 0, stream>>>(sim_softmax, fuse_weight, P);
  gnn_kernel<<<DRn, 256, 0, stream>>>(embed_table, gnn_W, gnn_b, fingerprints, adjacency, comp);
  conv1_stats_kernel<<<Bsz, 256, 0, stream>>>(rma, var, drug_id, P, conv1_w, conv1_b, com, p1);
  bn_reduce_kernel<<<1, 32, 0, stream>>>(p1, Bsz, C1, 1.0f / ((float)Bsz * (float)L1), bn1_g, bn1_b, ab1);
  conv2_stats_kernel<<<Bsz, 256, 0, stream>>>(com, conv1_w, conv1_b, ab1, conv2_w, conv2_b, out2, p2);
  bn_reduce_kernel<<<1, 32, 0, stream>>>(p2, Bsz, C2, 1.0f / ((float)Bsz * (float)L2), bn2_g, bn2_b, ab2);
  gene_final_kernel<<<Bsz, 256, 0, stream>>>(out2, ab2, lin_gene_w, lin_gene_b, comp, drug_id, conc);
  wmma_mlin_kernel<<<dim3((MH + 15) / 16, Bsz / 16), 32, 0, stream>>>(conc, mlin_w, mlin_b, Mt);
  mconv1_stats_kernel<<<Bsz, 64, 0, stream>>>(Mt, mconv1_w, mconv1_b, m1, pm1);
  bn_reduce_kernel<<<1, 32, 0, stream>>>(pm1, Bsz, C2, 1.0f / ((float)Bsz * (float)ML1), mbn1_g, mbn1_b, abm1);
  mconv2_stats_kernel<<<Bsz, 64, 0, stream>>>(m1, abm1, mconv2_w, mconv2_b, m2, pm2);
  bn_reduce_kernel<<<1, 32, 0, stream>>>(pm2, Bsz, C2, 1.0f / ((float)Bsz * (float)ML2), mbn2_g, mbn2_b, abm2);
  final_kernel<<<(Bsz + 255) / 256, 256, 0, stream>>>(m2, abm2, out_w, out_b, (float*)d_out);
}